// AttnDecoderRNN_85564338471302
// MI455X (gfx1250) — compile-verified
//
#include <hip/hip_runtime.h>
#include <hip/hip_bf16.h>
#include <math.h>

#define Hx 1024
#define Vx 50257
#define Lx 512
#define WSUM_CH 8
#define LSE_BLOCKS 64

typedef __attribute__((ext_vector_type(2))) float v2f;
typedef __attribute__((ext_vector_type(4))) float v4f;
typedef __attribute__((ext_vector_type(8))) float v8f;

#if defined(__gfx1250__) && __has_builtin(__builtin_amdgcn_wmma_f32_16x16x4_f32)
#define USE_WMMA_F32 1
#else
#define USE_WMMA_F32 0
#endif

// ---------------------------------------------------------------------------
// prep1: attn_in = [emb[input], h]   (2048 floats)
// ---------------------------------------------------------------------------
__global__ void prep1_kernel(const int* __restrict__ input,
                             const float* __restrict__ emb,
                             const float* __restrict__ hidden,
                             float* __restrict__ attn_in) {
    int i = blockIdx.x * blockDim.x + threadIdx.x;
    if (i < 2 * Hx) {
        attn_in[i] = (i < Hx) ? emb[(size_t)input[0] * Hx + i] : hidden[i - Hx];
    }
}

// ---------------------------------------------------------------------------
// prep2: comb_in = [embedded (== attn_in[0:H]), attn_applied]
// ---------------------------------------------------------------------------
__global__ void prep2_kernel(const float* __restrict__ attn_in,
                             const float* __restrict__ applied,
                             float* __restrict__ comb_in) {
    int i = blockIdx.x * blockDim.x + threadIdx.x;
    if (i < 2 * Hx) {
        comb_in[i] = (i < Hx) ? attn_in[i] : applied[i - Hx];
    }
}

// ---------------------------------------------------------------------------
// WMMA-based GEMV: y[r] = act( sum_k W[r][k] * x[k] + b[r] ),  W: MxK row-major.
// One wave (wave32) computes 16 consecutive rows using V_WMMA_F32_16X16X4_F32:
//   A (16x4 f32): VGPR0 = K0 (lanes 0-15) / K2 (lanes 16-31); VGPR1 = K1 / K3
//     -> lane loads global b64: W[row_base + (lane&15)][k0 + 2*(lane>>4) .. +1]
//   B (4x16 f32): same K half-split; x[k] broadcast into all 16 N columns, so
//     every column of D holds the same 16 dot products.
//   D (16x16 f32): VGPR j = row j (lanes 0-15) / row 8+j (lanes 16-31).
// Block = 256 threads = 8 waves = 128 rows. x staged in LDS with b128 copies
// (K is a multiple of 1024 here). Unroll 16 -> 16 outstanding global_load_b64
// per wave (4 KB in flight/wave; ~13 MB chip-wide on the 50257-row projection,
// covering the HBM latency-bandwidth product at 23.3 TB/s).
// act: 0 = none, 1 = relu.
// ---------------------------------------------------------------------------
__global__ void gemv_wmma_kernel(const float* __restrict__ W,
                                 const float* __restrict__ x,
                                 const float* __restrict__ bias,
                                 float* __restrict__ y,
                                 int M, int K, int act) {
    __shared__ float xs[2048];
    const int tid = threadIdx.x;
    // Vector staging: K/4 float4 elements, blockDim 256 -> 1..2 independent
    // b128 load/store pairs per thread (no serialized round trips).
    {
        const v4f* __restrict__ src = (const v4f*)x;
        v4f* __restrict__ dst = (v4f*)xs;
        const int n4 = K >> 2;
        for (int i = tid; i < n4; i += 256) dst[i] = src[i];
    }
    __syncthreads();

    const int wave = tid >> 5;          // wave32
    const int lane = tid & 31;
    const int half = lane >> 4;         // 0: lanes 0-15, 1: lanes 16-31
    const int mrow = lane & 15;
    const int row_base = (blockIdx.x * (blockDim.x >> 5) + wave) * 16;
    if (row_base >= M) return;          // wave-uniform exit (EXEC stays full)

    int r = row_base + mrow;
    if (r >= M) r = M - 1;              // clamp tail tile (results discarded)
    const float* wrow = W + (size_t)r * (size_t)K + (size_t)(2 * half);
    const float* xrow = xs + 2 * half;

#if USE_WMMA_F32
    v8f acc = {};
#else
    float acc_s = 0.f;
#endif

#pragma unroll 16
    for (int k0 = 0; k0 < K; k0 += 4) {
        v2f a  = *(const v2f*)(wrow + k0);   // global_load_b64
        v2f bm = *(const v2f*)(xrow + k0);   // ds_load_b64 (broadcast per half)
#if USE_WMMA_F32
        acc = __builtin_amdgcn_wmma_f32_16x16x4_f32(
            /*neg_a=*/false, a, /*neg_b=*/false, bm,
            /*c_mod=*/(short)0, acc, /*reuse_a=*/false, /*reuse_b=*/false);
#else
        acc_s += a.x * bm.x + a.y * bm.y;
#endif
    }

#if USE_WMMA_F32
    // Every N column of D is identical; lane 0 holds rows base+0..7 in acc[0..7],
    // lane 16 holds rows base+8..15.
    if (lane == 0 || lane == 16) {
        const int rb = row_base + (half ? 8 : 0);
#pragma unroll
        for (int j = 0; j < 8; ++j) {
            int rr = rb + j;
            if (rr < M) {
                float v = acc[j] + bias[rr];
                if (act == 1) v = fmaxf(v, 0.f);
                y[rr] = v;
            }
        }
    }
#else
    float other = __shfl_xor(acc_s, 16, 32);
    float tot = acc_s + other;
    if (lane < 16) {
        int rr = row_base + lane;
        if (rr < M) {
            float v = tot + bias[rr];
            if (act == 1) v = fmaxf(v, 0.f);
            y[rr] = v;
        }
    }
#endif
}

// ---------------------------------------------------------------------------
// softmax over L=512 scores (one block, 512 threads); also copies result to
// the attn_weights slot of d_out.
// ---------------------------------------------------------------------------
__global__ void softmax_L_kernel(float* __restrict__ v,
                                 float* __restrict__ out_copy) {
    __shared__ float sh[Lx];
    const int t = threadIdx.x;
    float x = v[t];
    sh[t] = x;
    __syncthreads();
    for (int off = Lx / 2; off > 0; off >>= 1) {
        if (t < off) sh[t] = fmaxf(sh[t], sh[t + off]);
        __syncthreads();
    }
    float mx = sh[0];
    __syncthreads();
    float e = expf(x - mx);
    sh[t] = e;
    __syncthreads();
    for (int off = Lx / 2; off > 0; off >>= 1) {
        if (t < off) sh[t] += sh[t + off];
        __syncthreads();
    }
    float r = e / sh[0];
    v[t] = r;
    out_copy[t] = r;
}

// ---------------------------------------------------------------------------
// attn_applied partials: part[chunk][c] = sum over one L-chunk of w[l]*enc[l][c]
// grid = (Hx/256, WSUM_CH); coalesced over columns.
// ---------------------------------------------------------------------------
__global__ void wsum_partial_kernel(const float* __restrict__ w,
                                    const float* __restrict__ enc,
                                    float* __restrict__ part) {
    const int c  = blockIdx.x * blockDim.x + threadIdx.x;
    const int l0 = blockIdx.y * (Lx / WSUM_CH);
    float s = 0.f;
    for (int l = l0; l < l0 + Lx / WSUM_CH; ++l) s += w[l] * enc[(size_t)l * Hx + c];
    part[(size_t)blockIdx.y * Hx + c] = s;
}

__global__ void wsum_reduce_kernel(const float* __restrict__ part,
                                   float* __restrict__ out) {
    const int c = blockIdx.x * blockDim.x + threadIdx.x;
    float s = 0.f;
#pragma unroll
    for (int j = 0; j < WSUM_CH; ++j) s += part[(size_t)j * Hx + c];
    out[c] = s;
}

// ---------------------------------------------------------------------------
// GRU gates (PyTorch layout r,z,n): h_new = (1-z)*n + z*h
// ---------------------------------------------------------------------------
__global__ void gru_kernel(const float* __restrict__ gx,
                           const float* __restrict__ gh,
                           const float* __restrict__ h,
                           float* __restrict__ hnew,
                           float* __restrict__ dout_h) {
    int i = blockIdx.x * blockDim.x + threadIdx.x;
    if (i < Hx) {
        float r = 1.f / (1.f + expf(-(gx[i] + gh[i])));
        float z = 1.f / (1.f + expf(-(gx[Hx + i] + gh[Hx + i])));
        float n = tanhf(gx[2 * Hx + i] + r * gh[2 * Hx + i]);
        float o = (1.f - z) * n + z * h[i];
        hnew[i] = o;
        dout_h[i] = o;
    }
}

// ---------------------------------------------------------------------------
// logsumexp, pass 1: per-block local max + local sum(exp(x - localmax)).
// 64 blocks x 256 threads; logits stay L2-resident between the two sweeps.
// ---------------------------------------------------------------------------
__global__ void lse_partial_kernel(const float* __restrict__ logits,
                                   float* __restrict__ pmax,
                                   float* __restrict__ psum) {
    __shared__ float sh[256];
    const int t = threadIdx.x;
    const int stride = LSE_BLOCKS * 256;
    float mx = -INFINITY;
    for (int i = blockIdx.x * 256 + t; i < Vx; i += stride) mx = fmaxf(mx, logits[i]);
    sh[t] = mx;
    __syncthreads();
    for (int off = 128; off > 0; off >>= 1) {
        if (t < off) sh[t] = fmaxf(sh[t], sh[t + off]);
        __syncthreads();
    }
    const float bmax = sh[0];
    __syncthreads();
    float s = 0.f;
    for (int i = blockIdx.x * 256 + t; i < Vx; i += stride) s += expf(logits[i] - bmax);
    sh[t] = s;
    __syncthreads();
    for (int off = 128; off > 0; off >>= 1) {
        if (t < off) sh[t] += sh[t + off];
        __syncthreads();
    }
    if (t == 0) { pmax[blockIdx.x] = bmax; psum[blockIdx.x] = sh[0]; }
}

// ---------------------------------------------------------------------------
// logsumexp, pass 2: combine 64 partials -> stats = {gmax, log(sum)}
// sum = sum_i psum[i] * exp(pmax[i] - gmax)
// ---------------------------------------------------------------------------
__global__ void lse_final_kernel(const float* __restrict__ pmax,
                                 const float* __restrict__ psum,
                                 float* __restrict__ stats) {
    __shared__ float sm[LSE_BLOCKS];
    __shared__ float ss[LSE_BLOCKS];
    const int t = threadIdx.x;   // LSE_BLOCKS threads
    float m = pmax[t];
    sm[t] = m;
    __syncthreads();
    for (int off = LSE_BLOCKS / 2; off > 0; off >>= 1) {
        if (t < off) sm[t] = fmaxf(sm[t], sm[t + off]);
        __syncthreads();
    }
    const float gmax = sm[0];
    __syncthreads();
    ss[t] = psum[t] * expf(m - gmax);
    __syncthreads();
    for (int off = LSE_BLOCKS / 2; off > 0; off >>= 1) {
        if (t < off) ss[t] += ss[t + off];
        __syncthreads();
    }
    if (t == 0) { stats[0] = gmax; stats[1] = logf(ss[0]); }
}

// ---------------------------------------------------------------------------
// logp[i] = logits[i] - max - log(sum)
// ---------------------------------------------------------------------------
__global__ void logp_kernel(const float* __restrict__ logits,
                            const float* __restrict__ stats,
                            float* __restrict__ out) {
    int i = blockIdx.x * blockDim.x + threadIdx.x;
    if (i < Vx) out[i] = logits[i] - stats[0] - stats[1];
}

// ---------------------------------------------------------------------------
extern "C" void kernel_launch(void* const* d_in, const int* in_sizes, int n_in,
                              void* d_out, int out_size, void* d_ws, size_t ws_size,
                              hipStream_t stream) {
    (void)in_sizes; (void)n_in; (void)out_size; (void)ws_size;

    const int*   input    = (const int*)  d_in[0];
    const float* hidden   = (const float*)d_in[1];   // (1,1,H)
    const float* enc      = (const float*)d_in[2];   // (L,H)
    const float* emb      = (const float*)d_in[3];   // (V,H)
    const float* attn_W   = (const float*)d_in[4];   // (L,2H)
    const float* attn_b   = (const float*)d_in[5];   // (L)
    const float* comb_W   = (const float*)d_in[6];   // (H,2H)
    const float* comb_b   = (const float*)d_in[7];   // (H)
    const float* gru_w_ih = (const float*)d_in[8];   // (3H,H)
    const float* gru_w_hh = (const float*)d_in[9];   // (3H,H)
    const float* gru_b_ih = (const float*)d_in[10];  // (3H)
    const float* gru_b_hh = (const float*)d_in[11];  // (3H)
    const float* out_W    = (const float*)d_in[12];  // (V,H)
    const float* out_b    = (const float*)d_in[13];  // (V)

    float* out = (float*)d_out;           // [logp(V) | h_new(H) | attn_w(L)]
    float* out_logp = out;
    float* out_h    = out + Vx;
    float* out_attn = out + Vx + Hx;

    float* ws        = (float*)d_ws;
    float* attn_in   = ws;                // 2048
    float* scores    = ws + 2048;         // 512
    float* applied   = ws + 2560;         // 1024
    float* comb_in   = ws + 3584;         // 2048
    float* xbuf      = ws + 5632;         // 1024
    float* gx        = ws + 6656;         // 3072
    float* gh        = ws + 9728;         // 3072
    float* hnew      = ws + 12800;        // 1024
    float* logits    = ws + 13824;        // 50257
    float* stats     = ws + 64084;        // 2
    float* wsum_part = ws + 64128;        // WSUM_CH * 1024 = 8192
    float* lse_pmax  = ws + 72320;        // 64
    float* lse_psum  = ws + 72384;        // 64

    // 1) attn_in = [emb[input], h]
    prep1_kernel<<<(2 * Hx + 255) / 256, 256, 0, stream>>>(input, emb, hidden, attn_in);
    // 2) scores = attn_in @ attn_W.T + attn_b          (512 x 2048 WMMA-GEMV)
    gemv_wmma_kernel<<<(Lx + 127) / 128, 256, 0, stream>>>(attn_W, attn_in, attn_b, scores, Lx, 2 * Hx, 0);
    // 3) softmax over 512 -> attn_weights (also to d_out)
    softmax_L_kernel<<<1, Lx, 0, stream>>>(scores, out_attn);
    // 4) attn_applied = attn_weights @ encoder_outputs (split-L partials + reduce)
    {
        dim3 g(Hx / 256, WSUM_CH);
        wsum_partial_kernel<<<g, 256, 0, stream>>>(scores, enc, wsum_part);
        wsum_reduce_kernel<<<Hx / 256, 256, 0, stream>>>(wsum_part, applied);
    }
    // 5) comb_in = [embedded, attn_applied]
    prep2_kernel<<<(2 * Hx + 255) / 256, 256, 0, stream>>>(attn_in, applied, comb_in);
    // 6) x = relu(comb_in @ comb_W.T + comb_b)          (1024 x 2048 WMMA-GEMV)
    gemv_wmma_kernel<<<(Hx + 127) / 128, 256, 0, stream>>>(comb_W, comb_in, comb_b, xbuf, Hx, 2 * Hx, 1);
    // 7) gx = x @ gru_w_ih.T + b_ih ; gh = h @ gru_w_hh.T + b_hh  (3072 x 1024 each)
    gemv_wmma_kernel<<<(3 * Hx + 127) / 128, 256, 0, stream>>>(gru_w_ih, xbuf,   gru_b_ih, gx, 3 * Hx, Hx, 0);
    gemv_wmma_kernel<<<(3 * Hx + 127) / 128, 256, 0, stream>>>(gru_w_hh, hidden, gru_b_hh, gh, 3 * Hx, Hx, 0);
    // 8) GRU gates -> h_new (also to d_out)
    gru_kernel<<<(Hx + 255) / 256, 256, 0, stream>>>(gx, gh, hidden, hnew, out_h);
    // 9) logits = h_new @ out_W.T + out_b               (50257 x 1024 WMMA-GEMV, ~206 MB)
    gemv_wmma_kernel<<<(Vx + 127) / 128, 256, 0, stream>>>(out_W, hnew, out_b, logits, Vx, Hx, 0);
    // 10) log-softmax: partial (64 blocks) + combine, then final write
    lse_partial_kernel<<<LSE_BLOCKS, 256, 0, stream>>>(logits, lse_pmax, lse_psum);
    lse_final_kernel<<<1, LSE_BLOCKS, 0, stream>>>(lse_pmax, lse_psum, stats);
    logp_kernel<<<(Vx + 255) / 256, 256, 0, stream>>>(logits, stats, out_logp);
}